// TurboQuantMSE_876173328527
// MI455X (gfx1250) — compile-verified
//
#include <hip/hip_runtime.h>

#define D_DIM 1024   // feature dim
#define N_ROWS 4096  // rows of x
#define KC 32        // K slice staged in LDS
#define BM 128       // workgroup tile M
#define BN 128       // workgroup tile N
#define APAD 4       // row stride = KC+APAD = 36 words: 16B-aligned rows + conflict-free

typedef float v2f __attribute__((ext_vector_type(2)));
typedef float v8f __attribute__((ext_vector_type(8)));
typedef int   tq_v4i __attribute__((vector_size(16)));

// ---------------------------------------------------------------------------
// CDNA5 async memory->LDS path (ASYNCcnt-tracked), with a safe fallback.
// Signature (from hipcc diagnostic): param0 = v4i* in global AS, param1 = LDS.
// ---------------------------------------------------------------------------
#if __has_builtin(__builtin_amdgcn_global_load_async_to_lds_b128) && \
    __has_builtin(__builtin_amdgcn_s_wait_asynccnt)
#define TQ_ASYNC 1
__device__ __forceinline__ void cp16_async(const void* g, void* l) {
    __builtin_amdgcn_global_load_async_to_lds_b128(
        (__attribute__((address_space(1))) tq_v4i*)(uintptr_t)g,
        (__attribute__((address_space(3))) tq_v4i*)(unsigned)(uintptr_t)l,
        /*imm_offset=*/0, /*cpol=*/0);
}
#define TQ_ASYNC_WAIT(n) __builtin_amdgcn_s_wait_asynccnt(n)
#else
#define TQ_ASYNC 0
__device__ __forceinline__ void cp16_async(const void* g, void* l) {
    *(float4*)l = *(const float4*)g;   // sync fallback; barrier handles ordering
}
#define TQ_ASYNC_WAIT(n) ((void)0)
#endif

// ---------------------------------------------------------------------------
// Kernel 1: y = x @ Pi^T, then 4-bit Lloyd-Max quantize -> int32 indices.
// Both LDS tiles row-major [128][36] so staging is contiguous 16B chunks.
// ---------------------------------------------------------------------------
__global__ __launch_bounds__(256)
void tq_rotate_quant(const float* __restrict__ x,
                     const float* __restrict__ Pi,
                     const float* __restrict__ cen,
                     int* __restrict__ idx_out) {
    __shared__ float Alds[2][BM][KC + APAD];   // A[m][k] = x[m0+m][kb+k]
    __shared__ float Blds[2][BN][KC + APAD];   // B[n][k] = Pi[n0+n][kb+k]

    const int tid   = threadIdx.x;
    const int lane  = tid & 31;
    const int wave  = tid >> 5;
    const int waveM = wave & 3;     // 4 waves along M (32 rows each)
    const int waveN = wave >> 2;    // 2 waves along N (64 cols each)
    const int m0 = blockIdx.y * BM;
    const int n0 = blockIdx.x * BN;

    const int halfsel = lane >> 4;
    const int l16     = lane & 15;

    // Sorted centroids: nearest index = count of midpoints below value (SGPRs).
    float mid[15];
#pragma unroll
    for (int j = 0; j < 15; ++j) mid[j] = 0.5f * (cen[j] + cen[j + 1]);

    v8f acc[2][4];
#pragma unroll
    for (int mi = 0; mi < 2; ++mi)
#pragma unroll
        for (int ni = 0; ni < 4; ++ni)
#pragma unroll
            for (int e = 0; e < 8; ++e) acc[mi][ni][e] = 0.0f;

    const int c4 = (tid & 7) * 4;   // 0..28 (k chunk)
    const int r  = tid >> 3;        // 0..31 (row group)

    auto stage = [&](int buf, int kb) {
#pragma unroll
        for (int rr = 0; rr < 4; ++rr) {
            const int row = r + rr * 32;
            cp16_async(&x[(size_t)(m0 + row) * D_DIM + kb + c4],
                       &Alds[buf][row][c4]);
            cp16_async(&Pi[(size_t)(n0 + row) * D_DIM + kb + c4],
                       &Blds[buf][row][c4]);
        }
    };

    stage(0, 0);   // prologue: 8 async copies in flight

    for (int kb = 0; kb < D_DIM; kb += KC) {
        const int cur = (kb / KC) & 1;
        if (kb + KC < D_DIM) {
            stage(cur ^ 1, kb + KC);   // issue next slice (16 in flight)
            TQ_ASYNC_WAIT(8);          // in-order: current slice resident
        } else {
            TQ_ASYNC_WAIT(0);
        }
        __syncthreads();

#pragma unroll
        for (int ks = 0; ks < KC; ks += 4) {
            const int ka = ks + 2 * halfsel;   // ISA A/B fragment layout
            v2f a[2], b[4];
#pragma unroll
            for (int mi = 0; mi < 2; ++mi)
                a[mi] = *(const v2f*)&Alds[cur][waveM * 32 + mi * 16 + l16][ka];
#pragma unroll
            for (int ni = 0; ni < 4; ++ni)
                b[ni] = *(const v2f*)&Blds[cur][waveN * 64 + ni * 16 + l16][ka];
#pragma unroll
            for (int mi = 0; mi < 2; ++mi)
#pragma unroll
                for (int ni = 0; ni < 4; ++ni)
                    acc[mi][ni] = __builtin_amdgcn_wmma_f32_16x16x4_f32(
                        false, a[mi], false, b[ni], (short)0, acc[mi][ni],
                        false, false);
        }
        __syncthreads();
    }

    // Quantize + store indices (C/D layout: row = vgpr + 8*(lane/16))
#pragma unroll
    for (int mi = 0; mi < 2; ++mi)
#pragma unroll
        for (int ni = 0; ni < 4; ++ni)
#pragma unroll
            for (int e = 0; e < 8; ++e) {
                const int gm = m0 + waveM * 32 + mi * 16 + e + 8 * halfsel;
                const int gn = n0 + waveN * 64 + ni * 16 + l16;
                const float v = acc[mi][ni][e];
                int q = 0;
#pragma unroll
                for (int j = 0; j < 15; ++j) q += (v > mid[j]) ? 1 : 0;
                idx_out[(size_t)gm * D_DIM + gn] = q;
            }
}

// ---------------------------------------------------------------------------
// Kernel 2: x_hat = centroids[idx] @ Pi. A dequantized on the fly (16-entry
// LDS table) while staging; B tile async-copied; both double-buffered.
// ---------------------------------------------------------------------------
__global__ __launch_bounds__(256)
void tq_dequant_unrotate(const int* __restrict__ idx_in,
                         const float* __restrict__ Pi,
                         const float* __restrict__ cen,
                         float* __restrict__ xhat) {
    __shared__ float Alds[2][BM][KC + APAD];   // dequantized centroids
    __shared__ float Blds[2][KC][BN];          // B[k][n] = Pi[kb+k][n0+n]
    __shared__ float ctab[16];

    const int tid   = threadIdx.x;
    const int lane  = tid & 31;
    const int wave  = tid >> 5;
    const int waveM = wave & 3;
    const int waveN = wave >> 2;
    const int m0 = blockIdx.y * BM;
    const int n0 = blockIdx.x * BN;

    const int halfsel = lane >> 4;
    const int l16     = lane & 15;

    if (tid < 16) ctab[tid] = cen[tid];
    __syncthreads();

    v8f acc[2][4];
#pragma unroll
    for (int mi = 0; mi < 2; ++mi)
#pragma unroll
        for (int ni = 0; ni < 4; ++ni)
#pragma unroll
            for (int e = 0; e < 8; ++e) acc[mi][ni][e] = 0.0f;

    const int c4 = (tid & 7) * 4;
    const int r  = tid >> 3;
    const int n4 = (tid & 31) * 4;  // 0..124 (n chunk)
    const int kq = tid >> 5;        // 0..7   (k row group)

    auto stageB = [&](int buf, int kb) {
#pragma unroll
        for (int kk = 0; kk < 4; ++kk) {
            const int krow = kq + kk * 8;
            cp16_async(&Pi[(size_t)(kb + krow) * D_DIM + n0 + n4],
                       &Blds[buf][krow][n4]);
        }
    };
    auto stageA = [&](int buf, int kb) {
#pragma unroll
        for (int rr = 0; rr < 4; ++rr) {
            const int row = r + rr * 32;
            int4 q = *(const int4*)&idx_in[(size_t)(m0 + row) * D_DIM + kb + c4];
            Alds[buf][row][c4 + 0] = ctab[q.x & 15];
            Alds[buf][row][c4 + 1] = ctab[q.y & 15];
            Alds[buf][row][c4 + 2] = ctab[q.z & 15];
            Alds[buf][row][c4 + 3] = ctab[q.w & 15];
        }
    };

    stageB(0, 0);
    stageA(0, 0);

    for (int kb = 0; kb < D_DIM; kb += KC) {
        const int cur = (kb / KC) & 1;
        if (kb + KC < D_DIM) {
            stageB(cur ^ 1, kb + KC);
            stageA(cur ^ 1, kb + KC);
            TQ_ASYNC_WAIT(4);
        } else {
            TQ_ASYNC_WAIT(0);
        }
        __syncthreads();

#pragma unroll
        for (int ks = 0; ks < KC; ks += 4) {
            const int ka = ks + 2 * halfsel;
            v2f a[2], b[4];
#pragma unroll
            for (int mi = 0; mi < 2; ++mi)
                a[mi] = *(const v2f*)&Alds[cur][waveM * 32 + mi * 16 + l16][ka];
#pragma unroll
            for (int ni = 0; ni < 4; ++ni) {
                const int n = waveN * 64 + ni * 16 + l16;
                v2f bv;
                bv[0] = Blds[cur][ka][n];
                bv[1] = Blds[cur][ka + 1][n];
                b[ni] = bv;
            }
#pragma unroll
            for (int mi = 0; mi < 2; ++mi)
#pragma unroll
                for (int ni = 0; ni < 4; ++ni)
                    acc[mi][ni] = __builtin_amdgcn_wmma_f32_16x16x4_f32(
                        false, a[mi], false, b[ni], (short)0, acc[mi][ni],
                        false, false);
        }
        __syncthreads();
    }

#pragma unroll
    for (int mi = 0; mi < 2; ++mi)
#pragma unroll
        for (int ni = 0; ni < 4; ++ni)
#pragma unroll
            for (int e = 0; e < 8; ++e) {
                const int gm = m0 + waveM * 32 + mi * 16 + e + 8 * halfsel;
                const int gn = n0 + waveN * 64 + ni * 16 + l16;
                xhat[(size_t)gm * D_DIM + gn] = acc[mi][ni][e];
            }
}

// ---------------------------------------------------------------------------
extern "C" void kernel_launch(void* const* d_in, const int* in_sizes, int n_in,
                              void* d_out, int out_size, void* d_ws, size_t ws_size,
                              hipStream_t stream) {
    const float* x   = (const float*)d_in[0];   // [N, D] f32
    const float* Pi  = (const float*)d_in[1];   // [D, D] f32
    const float* cen = (const float*)d_in[2];   // [16]   f32

    float* xhat = (float*)d_out;                          // first N*D floats
    int*   idx  = (int*)d_out + (size_t)N_ROWS * D_DIM;   // next N*D int32

    dim3 grid(D_DIM / BN, N_ROWS / BM);   // 8 x 32 workgroups
    dim3 block(256);

    tq_rotate_quant<<<grid, block, 0, stream>>>(x, Pi, cen, idx);
    tq_dequant_unrotate<<<grid, block, 0, stream>>>(idx, Pi, cen, xhat);
}